// RNNDecoder_61177514164512
// MI455X (gfx1250) — compile-verified
//
#include <hip/hip_runtime.h>

typedef float v2f __attribute__((ext_vector_type(2)));
typedef float v8f __attribute__((ext_vector_type(8)));

constexpr int B_    = 32;
constexpr int T_    = 2048;
constexpr int IN_   = 576;
constexpr int H_    = 60;      // hidden per direction
constexpr int G_    = 180;     // 3*H gates
constexpr int HID2_ = 120;     // 2*H bidirectional concat
constexpr int ROWS  = B_ * T_; // 65536 (row index r = b*T + t everywhere)
constexpr int WPAD  = 61;      // LDS row pitch for W_hh (gcd(61,64)=1 -> conflict-free)

#define WMMA_F32(A, Bv, C) \
    __builtin_amdgcn_wmma_f32_16x16x4_f32(false, (A), false, (Bv), (short)0, (C), false, false)

// ---------------------------------------------------------------------------
// gx[r][g] = sum_k in[r][k] * w[dir][g][k] + bias[dir][g]
// WMMA f32 16x16x4. Each wave: 4 M-tiles x 4 N-tiles (16 v8f accumulators),
// fragments software-pipelined one k-step ahead; 8 loads feed 16 WMMAs.
// grid = (ROWS/256, 3, 2 dirs), block = (32, 4 waves)
// ---------------------------------------------------------------------------
template <int K>
__global__ void __launch_bounds__(128)
gru_gemm_gx(const float* __restrict__ in, const float* __restrict__ w,
            const float* __restrict__ bias, float* __restrict__ gx)
{
    const int lane  = threadIdx.x;                  // 0..31
    const int wave  = threadIdx.y;                  // 0..3
    const int dir   = blockIdx.z;
    const int mT    = (blockIdx.x * 4 + wave) * 4;  // first of 4 M-tiles
    const int gBase = blockIdx.y * 64;

    const int m     = lane & 15;
    const int kHalf = lane >> 4;                    // 0: K=k0..k0+1, 1: K=k0+2..k0+3
    const int n     = lane & 15;

    const float* wd   = w    + (size_t)dir * G_ * K;
    const float* bd   = bias + dir * G_;
    float*       outd = gx   + (size_t)dir * (size_t)ROWS * G_;

    // A fragment base: rows mT*16+m .. ; tiles i at compile-time offset i*16*K
    const float* aB = in + (size_t)(mT * 16 + m) * K + kHalf * 2;

    // B fragment sources: 4 N-tiles; clamp OOB gate columns to row 0 (stores guarded)
    int gcol[4];
    const float* wb[4];
    #pragma unroll
    for (int j = 0; j < 4; ++j) {
        gcol[j] = gBase + j * 16 + n;
        wb[j]   = wd + (size_t)((gcol[j] < G_) ? gcol[j] : 0) * K + kHalf * 2;
    }

    v8f acc[4][4];
    #pragma unroll
    for (int i = 0; i < 4; ++i)
        #pragma unroll
        for (int j = 0; j < 4; ++j) acc[i][j] = (v8f){};

    // prologue: fragments for k0 = 0
    v2f aC[4], bC[4];
    #pragma unroll
    for (int i = 0; i < 4; ++i) aC[i] = *(const v2f*)(aB + (size_t)i * 16 * K);
    #pragma unroll
    for (int j = 0; j < 4; ++j) bC[j] = *(const v2f*)(wb[j]);

    #pragma unroll 4
    for (int k0 = 4; k0 < K; k0 += 4) {
        // issue next k-step's loads before consuming current fragments
        v2f aN[4], bN[4];
        #pragma unroll
        for (int i = 0; i < 4; ++i) aN[i] = *(const v2f*)(aB + (size_t)i * 16 * K + k0);
        #pragma unroll
        for (int j = 0; j < 4; ++j) bN[j] = *(const v2f*)(wb[j] + k0);

        // sparse prefetch of upcoming A cachelines (one per 32 k-floats)
        if ((k0 % 32) == 4) {
            #pragma unroll
            for (int i = 0; i < 4; ++i)
                __builtin_prefetch(aB + (size_t)i * 16 * K + k0 + 124, 0, 1);
        }

        #pragma unroll
        for (int i = 0; i < 4; ++i)
            #pragma unroll
            for (int j = 0; j < 4; ++j)
                acc[i][j] = WMMA_F32(aC[i], bC[j], acc[i][j]);

        #pragma unroll
        for (int i = 0; i < 4; ++i) aC[i] = aN[i];
        #pragma unroll
        for (int j = 0; j < 4; ++j) bC[j] = bN[j];
    }
    // epilogue: last k-step
    #pragma unroll
    for (int i = 0; i < 4; ++i)
        #pragma unroll
        for (int j = 0; j < 4; ++j)
            acc[i][j] = WMMA_F32(aC[i], bC[j], acc[i][j]);

    // C/D layout: VGPR jj -> M = jj + (lane>=16 ? 8 : 0), N = lane&15
    float bv[4];
    #pragma unroll
    for (int j = 0; j < 4; ++j) bv[j] = (gcol[j] < G_) ? bd[gcol[j]] : 0.f;

    #pragma unroll
    for (int i = 0; i < 4; ++i) {
        const int mRow = mT * 16 + i * 16 + kHalf * 8;
        #pragma unroll
        for (int jj = 0; jj < 8; ++jj) {
            size_t row = (size_t)(mRow + jj) * G_;
            #pragma unroll
            for (int j = 0; j < 4; ++j)
                if (gcol[j] < G_) outd[row + gcol[j]] = acc[i][j][jj] + bv[j];
        }
    }
}

// ---------------------------------------------------------------------------
// Sequential GRU recurrence; one block per (batch b, direction).
// W_hh (padded rows, bank-conflict-free) / bias / h / gh in LDS.
// gx gate loads for step t+1 issued at the top of step t. grid=(32,2), blk=192.
// ---------------------------------------------------------------------------
__global__ void __launch_bounds__(192)
gru_recurrence(const float* __restrict__ gx, const float* __restrict__ w_hh,
               const float* __restrict__ b_hh, float* __restrict__ h_out)
{
    __shared__ float w_s[G_ * WPAD];   // 180*61*4 = 43.9 KB
    __shared__ float bias_s[G_];
    __shared__ float h_s[H_];
    __shared__ float gh_s[G_];

    const int b   = blockIdx.x;
    const int dir = blockIdx.y;
    const int tid = threadIdx.x;

    const float* wd  = w_hh + (size_t)dir * G_ * H_;
    const float* bd  = b_hh + dir * G_;
    const float* gxd = gx   + (size_t)dir * (size_t)ROWS * G_;

    for (int i = tid; i < G_ * H_; i += 192) {
        int g = i / H_, k = i - g * H_;
        w_s[g * WPAD + k] = wd[i];
    }
    for (int i = tid; i < G_; i += 192) bias_s[i] = bd[i];
    if (tid < H_) h_s[tid] = 0.f;

    // preload gate inputs for t = 0
    const int t0 = dir ? (T_ - 1) : 0;
    float xr = 0.f, xz = 0.f, xn = 0.f;
    if (tid < H_) {
        const float* g0 = gxd + ((size_t)b * T_ + t0) * G_;
        xr = g0[tid]; xz = g0[H_ + tid]; xn = g0[2 * H_ + tid];
    }
    __syncthreads();

    for (int t = 0; t < T_; ++t) {
        const int tt = dir ? (T_ - 1 - t) : t;

        // issue next step's gate loads early (consumed next iteration)
        float nxr = 0.f, nxz = 0.f, nxn = 0.f;
        if (t + 1 < T_ && tid < H_) {
            const int tn = dir ? (T_ - 2 - t) : (t + 1);
            const float* gn = gxd + ((size_t)b * T_ + tn) * G_;
            nxr = gn[tid]; nxz = gn[H_ + tid]; nxn = gn[2 * H_ + tid];
        }

        if (tid < G_) {
            float acc = bias_s[tid];
            const float* wr = &w_s[tid * WPAD];
            #pragma unroll 10
            for (int k = 0; k < H_; ++k) acc += h_s[k] * wr[k];
            gh_s[tid] = acc;
        }
        __syncthreads();

        if (tid < H_) {
            float hr = gh_s[tid], hz = gh_s[H_ + tid], hn = gh_s[2 * H_ + tid];
            float rg = 1.f / (1.f + __expf(-(xr + hr)));
            float zg = 1.f / (1.f + __expf(-(xz + hz)));
            float ng = tanhf(xn + rg * hn);
            float hv = (1.f - zg) * ng + zg * h_s[tid];
            h_s[tid] = hv;
            h_out[((size_t)b * T_ + tt) * HID2_ + dir * H_ + tid] = hv;
        }
        xr = nxr; xz = nxz; xn = nxn;
        __syncthreads();
    }
}

// ---------------------------------------------------------------------------
// out[r][o] = sum_f h[r][f] * fc_w[o][f] + fc_b[o]  (r = b*T+t matches output)
// ---------------------------------------------------------------------------
__global__ void __launch_bounds__(256)
fc_head(const float* __restrict__ h, const float* __restrict__ fw,
        const float* __restrict__ fb, float* __restrict__ out)
{
    int idx = blockIdx.x * 256 + threadIdx.x;
    if (idx >= ROWS * 5) return;
    int r = idx / 5, o = idx - r * 5;
    const float* hr = h + (size_t)r * HID2_;
    const float* wr = fw + o * HID2_;
    float acc = fb[o];
    #pragma unroll 8
    for (int f = 0; f < HID2_; ++f) acc += hr[f] * wr[f];
    out[idx] = acc;
}

// ---------------------------------------------------------------------------
extern "C" void kernel_launch(void* const* d_in, const int* in_sizes, int n_in,
                              void* d_out, int out_size, void* d_ws, size_t ws_size,
                              hipStream_t stream)
{
    (void)in_sizes; (void)n_in; (void)out_size; (void)ws_size;

    const float* x     = (const float*)d_in[0];
    const float* w_ih0 = (const float*)d_in[1];   // (2,180,576)
    const float* w_hh0 = (const float*)d_in[2];   // (2,180,60)
    const float* b_ih0 = (const float*)d_in[3];   // (2,180)
    const float* b_hh0 = (const float*)d_in[4];   // (2,180)
    const float* w_ih  = (const float*)d_in[5];   // (2,2,180,120)
    const float* w_hh  = (const float*)d_in[6];   // (2,2,180,60)
    const float* b_ih  = (const float*)d_in[7];   // (2,2,180)
    const float* b_hh  = (const float*)d_in[8];   // (2,2,180)
    const float* fc_w  = (const float*)d_in[9];   // (5,120)
    const float* fc_b  = (const float*)d_in[10];  // (5,)
    float* out = (float*)d_out;

    // Workspace: gx[2][ROWS][180] + hA[ROWS][120] + hB[ROWS][120]  (~157 MB)
    float* gx = (float*)d_ws;
    float* hA = gx + (size_t)2 * ROWS * G_;
    float* hB = hA + (size_t)ROWS * HID2_;

    dim3 gGrid(ROWS / 256, 3, 2), gBlk(32, 4);
    dim3 rGrid(B_, 2), rBlk(192);

    // layer 0
    gru_gemm_gx<IN_><<<gGrid, gBlk, 0, stream>>>(x, w_ih0, b_ih0, gx);
    gru_recurrence<<<rGrid, rBlk, 0, stream>>>(gx, w_hh0, b_hh0, hA);
    // layer 1
    gru_gemm_gx<HID2_><<<gGrid, gBlk, 0, stream>>>(hA, w_ih, b_ih, gx);
    gru_recurrence<<<rGrid, rBlk, 0, stream>>>(gx, w_hh, b_hh, hB);
    // layer 2
    gru_gemm_gx<HID2_><<<gGrid, gBlk, 0, stream>>>(hB, w_ih + 2 * G_ * HID2_, b_ih + 2 * G_, gx);
    gru_recurrence<<<rGrid, rBlk, 0, stream>>>(gx, w_hh + 2 * G_ * H_, b_hh + 2 * G_, hA);
    // FC head
    fc_head<<<dim3((ROWS * 5 + 255) / 256), dim3(256), 0, stream>>>(hA, fc_w, fc_b, out);
}